// XGate_72773925863428
// MI455X (gfx1250) — compile-verified
//
#include <hip/hip_runtime.h>
#include <hip/hip_bf16.h>

typedef float v2f __attribute__((ext_vector_type(2)));
typedef float v8f __attribute__((ext_vector_type(8)));

// out[o*D*S + r*S + c] = sum_i M[r,i] * x[o*D*S + i*S + c]
// with S = N / D^(index+1), o in [0, D^index), c in [0, S).
// One wave handles one (o, 16-column) tile via V_WMMA_F32_16X16X4_F32.
__global__ __launch_bounds__(256) void
xgate_wmma_kernel(const float* __restrict__ x,
                  const float* __restrict__ M,
                  const int* __restrict__ pIndex,
                  const int* __restrict__ pD,
                  float* __restrict__ out,
                  int N) {
    const int lane        = threadIdx.x & 31;
    const int wavesPerBlk = blockDim.x >> 5;
    const int waveId      = blockIdx.x * wavesPerBlk + (threadIdx.x >> 5);

    const int D     = pD[0];        // qudit dimension (2 here)
    const int index = pIndex[0];    // which qudit M acts on (0 here)

    // inner stride S = N / D^(index+1); outer = D^index
    int S = N;
    for (int i = 0; i <= index; ++i) S /= D;
    const int outer         = N / (D * S);
    const int tilesPerOuter = (S + 15) >> 4;
    const int numTiles      = outer * tilesPerOuter;

    // Wave-uniform exit: EXEC remains all-1s for waves that reach the WMMA.
    if (waveId >= numTiles) return;

    const int o    = waveId / tilesPerOuter;
    const int tile = waveId - o * tilesPerOuter;
    const int col0 = tile << 4;

    const float* __restrict__ xblk = x   + (size_t)o * D * S;
    float*       __restrict__ oblk = out + (size_t)o * D * S;

    // ---- A matrix: M zero-padded to 16x4 ----
    // lane l: row m = l&15 ; VGPR0/1 hold K = {0,1} (lanes 0-15) or {2,3} (lanes 16-31)
    const int m  = lane & 15;
    const int k0 = (lane >> 4) << 1;          // 0 or 2
    v2f a;
    a.x = (m < D && (k0 + 0) < D) ? M[m * D + (k0 + 0)] : 0.0f;
    a.y = (m < D && (k0 + 1) < D) ? M[m * D + (k0 + 1)] : 0.0f;

    // ---- B matrix: 4x16 tile of x (rows >= D are zero) ----
    // lane l: col n = l&15 ; VGPR0/1 hold K = {0,1} (lanes 0-15) or {2,3} (lanes 16-31)
    const int c = col0 + (lane & 15);
    const bool cOk = (c < S);
    v2f b;
    b.x = (cOk && (k0 + 0) < D) ? xblk[(size_t)(k0 + 0) * S + c] : 0.0f;
    b.y = (cOk && (k0 + 1) < D) ? xblk[(size_t)(k0 + 1) * S + c] : 0.0f;

    // ---- D = A x B + 0 : exact in fp32 (M entries are 0/1) ----
    v8f acc = {};
    acc = __builtin_amdgcn_wmma_f32_16x16x4_f32(
        /*neg_a=*/false, a, /*neg_b=*/false, b,
        /*c_mod=*/(short)0, acc, /*reuse_a=*/false, /*reuse_b=*/false);

    // ---- Store rows 0..D-1: D-matrix VGPR r, lanes 0-15 hold row r, col = lane ----
    if (lane < 16 && cOk) {
#pragma unroll
        for (int r = 0; r < 8; ++r) {
            if (r < D) oblk[(size_t)r * S + c] = acc[r];
        }
    }
}

extern "C" void kernel_launch(void* const* d_in, const int* in_sizes, int n_in,
                              void* d_out, int out_size, void* d_ws, size_t ws_size,
                              hipStream_t stream) {
    const float* x    = (const float*)d_in[0];
    const float* M    = (const float*)d_in[1];
    const int* pIndex = (const int*)d_in[2];
    const int* pD     = (const int*)d_in[3];
    float* out        = (float*)d_out;

    const int N = in_sizes[0];                  // 8192

    // Host knows D from M's element count (D*D), but not `index` (device
    // scalar). Worst case over index: numTiles <= N / D. Over-provision waves;
    // extras exit wave-uniformly.
    int D = 1;
    while (D * D < in_sizes[1]) ++D;            // D = 2

    const int maxWaves  = (N + D - 1) / D;      // 4096
    const int threads   = 256;                  // 8 waves per block (wave32)
    const int wavesPerB = threads / 32;
    const int blocks    = (maxWaves + wavesPerB - 1) / wavesPerB;

    xgate_wmma_kernel<<<blocks, threads, 0, stream>>>(x, M, pIndex, pD, out, N);
}